// SWM_FPRM_Loss_28063316313019
// MI455X (gfx1250) — compile-verified
//
#include <hip/hip_runtime.h>
#include <hip/hip_bf16.h>
#include <math.h>

// ---------------- problem constants (from reference) ----------------
#define B_SAMPLES 32
#define N_PER (384 * 384 * 2)          // 294912 elements per sample
#define RATIO 3                        // NEG_POS_RATIO
#define ALPHA 1.0f

// ---------------- tiling ----------------
#define EPB 4096                       // elements per block in streaming passes
#define BLOCKS_PER_SAMPLE (N_PER / EPB)  // 72
#define NBIN1 2048                     // bits[31:21]
#define NBIN2 2048                     // bits[20:10]
#define NBIN3 1024                     // bits[9:0]

typedef __attribute__((ext_vector_type(2))) float v2f;
typedef __attribute__((ext_vector_type(8))) float v8f;

// ---------------- workspace layout ----------------
struct Ws {
  float pos_loss[B_SAMPLES];
  float mse_sum;
  float sum_above[B_SAMPLES];
  int   rem[B_SAMPLES];
  unsigned b1[B_SAMPLES];
  unsigned b2[B_SAMPLES];
  float per_sample[B_SAMPLES];
  unsigned h1_cnt[B_SAMPLES][NBIN1];
  float    h1_sum[B_SAMPLES][NBIN1];
  unsigned h2_cnt[B_SAMPLES][NBIN2];
  float    h2_sum[B_SAMPLES][NBIN2];
  unsigned h3_cnt[B_SAMPLES][NBIN3];
  float    h3_sum[B_SAMPLES][NBIN3];
};

// ---------------- WMMA wave32 all-reduce (exact f32) ----------------
// Stage 1: A[m][0]=x(lane m), A[m][2]=x(lane m+16); B = ones 4x16.
//   D[m][n] = x_m + x_{m+16} (row-sum, replicated over n).
//   Per-lane add of D's 8 VGPRs -> lanes 0-15 hold S_lo, lanes 16-31 hold S_hi.
// Stage 2: same trick merges S_lo + S_hi -> total broadcast to all lanes.
// Requires full EXEC (call only with all lanes of the wave active).
__device__ __forceinline__ float wave32_allreduce_wmma(float x) {
  v2f a;   a[0] = x;    a[1] = 0.0f;
  v2f one; one[0] = 1.0f; one[1] = 1.0f;
  v8f c = {};
  v8f d = __builtin_amdgcn_wmma_f32_16x16x4_f32(
      /*neg_a=*/false, a, /*neg_b=*/false, one,
      /*c_mod=*/(short)0, c, /*reuse_a=*/false, /*reuse_b=*/false);
  float s = d[0] + d[1] + d[2] + d[3] + d[4] + d[5] + d[6] + d[7];
  v2f a2; a2[0] = s; a2[1] = 0.0f;
  v8f d2 = __builtin_amdgcn_wmma_f32_16x16x4_f32(
      false, a2, false, one, (short)0, c, false, false);
  return d2[0];
}

// ---------------- kernels ----------------
__global__ void k_init(unsigned* ws, int nwords) {
  int i = blockIdx.x * blockDim.x + threadIdx.x;
  int stride = gridDim.x * blockDim.x;
  for (; i < nwords; i += stride) ws[i] = 0u;
}

// Pass 1: stream y/out/w once; per-sample LDS histogram of neg (bits[31:21]),
// plus block-level pos_loss / mse partials reduced through WMMA all-reduce.
__global__ void k_pass1(const float* __restrict__ y, const float* __restrict__ o,
                        const float* __restrict__ w, Ws* ws) {
  __shared__ unsigned s_cnt[NBIN1];
  __shared__ float    s_sum[NBIN1];
  __shared__ float    s_red[16];
  const int b   = blockIdx.x / BLOCKS_PER_SAMPLE;
  const int blk = blockIdx.x % BLOCKS_PER_SAMPLE;
  const size_t base = (size_t)b * N_PER + (size_t)blk * EPB;

  for (int i = threadIdx.x; i < NBIN1; i += 256) { s_cnt[i] = 0u; s_sum[i] = 0.0f; }
  __syncthreads();

  float mse_acc = 0.0f, pos_acc = 0.0f;
  for (int i = threadIdx.x; i < EPB; i += 256) {
    size_t idx = base + i;
    __builtin_prefetch(&y[idx + 256], 0, 0);   // global_prefetch_b8
    __builtin_prefetch(&o[idx + 256], 0, 0);
    float yy = y[idx], oo = o[idx], ww = w[idx];
    float df = oo - yy;
    float mse = df * df;
    mse_acc += mse;
    bool pos = (ww > 0.0f);
    pos_acc += pos ? (ww * mse) : 0.0f;
    float neg = ((oo > 0.0f) && !pos) ? mse : 0.0f;
    unsigned bits = __float_as_uint(neg);
    unsigned bin  = bits >> 21;
    atomicAdd(&s_cnt[bin], 1u);
    atomicAdd(&s_sum[bin], neg);
  }
  __syncthreads();

  for (int i = threadIdx.x; i < NBIN1; i += 256) {
    unsigned c = s_cnt[i];
    if (c) {
      atomicAdd(&ws->h1_cnt[b][i], c);
      atomicAdd(&ws->h1_sum[b][i], s_sum[i]);
    }
  }

  // wave-level WMMA all-reduce (full EXEC: all 256 threads alive)
  float wm = wave32_allreduce_wmma(mse_acc);
  float wp = wave32_allreduce_wmma(pos_acc);
  int wid = threadIdx.x >> 5, lid = threadIdx.x & 31;
  if (lid == 0) { s_red[wid] = wm; s_red[8 + wid] = wp; }
  __syncthreads();
  // second stage: every wave runs it (uniform, full EXEC); only wave0 data is real
  float vm = (threadIdx.x < 8) ? s_red[threadIdx.x]     : 0.0f;
  float vp = (threadIdx.x < 8) ? s_red[8 + threadIdx.x] : 0.0f;
  float tm = wave32_allreduce_wmma(vm);
  float tp = wave32_allreduce_wmma(vp);
  if (threadIdx.x == 0) {
    atomicAdd(&ws->mse_sum, tm);
    atomicAdd(&ws->pos_loss[b], tp);
  }
}

__global__ void k_scan1(const float* __restrict__ total_size, Ws* ws) {
  int b = blockIdx.x;
  if (threadIdx.x != 0) return;
  float ts = total_size[b];
  long long kk = (long long)floorf(ts) * RATIO;
  if (kk < 0) kk = 0;
  if (kk > N_PER) kk = N_PER;
  long long cum = 0;
  float sumAb = 0.0f;
  unsigned bsel = 0; int rem = 0; bool found = false;
  for (int bin = NBIN1 - 1; bin >= 0 && !found; --bin) {
    unsigned c = ws->h1_cnt[b][bin];
    if (cum + (long long)c >= kk) { bsel = (unsigned)bin; rem = (int)(kk - cum); found = true; }
    else { cum += c; sumAb += ws->h1_sum[b][bin]; }
  }
  ws->b1[b] = bsel; ws->rem[b] = rem; ws->sum_above[b] = sumAb;
}

// Pass 2: re-stream inputs (L2-resident after pass 1: 113MB < 192MB L2);
// histogram bits[20:10] for elements in the boundary top-bin.
__global__ void k_pass2(const float* __restrict__ y, const float* __restrict__ o,
                        const float* __restrict__ w, Ws* ws) {
  __shared__ unsigned s_cnt[NBIN2];
  __shared__ float    s_sum[NBIN2];
  const int b   = blockIdx.x / BLOCKS_PER_SAMPLE;
  const int blk = blockIdx.x % BLOCKS_PER_SAMPLE;
  const size_t base = (size_t)b * N_PER + (size_t)blk * EPB;
  const unsigned target = ws->b1[b];

  for (int i = threadIdx.x; i < NBIN2; i += 256) { s_cnt[i] = 0u; s_sum[i] = 0.0f; }
  __syncthreads();

  for (int i = threadIdx.x; i < EPB; i += 256) {
    size_t idx = base + i;
    float yy = y[idx], oo = o[idx], ww = w[idx];
    float df = oo - yy;
    float mse = df * df;
    float neg = ((oo > 0.0f) && !(ww > 0.0f)) ? mse : 0.0f;
    unsigned bits = __float_as_uint(neg);
    if ((bits >> 21) == target) {
      unsigned bin = (bits >> 10) & 0x7FFu;
      atomicAdd(&s_cnt[bin], 1u);
      atomicAdd(&s_sum[bin], neg);
    }
  }
  __syncthreads();
  for (int i = threadIdx.x; i < NBIN2; i += 256) {
    unsigned c = s_cnt[i];
    if (c) {
      atomicAdd(&ws->h2_cnt[b][i], c);
      atomicAdd(&ws->h2_sum[b][i], s_sum[i]);
    }
  }
}

__global__ void k_scan2(Ws* ws) {
  int b = blockIdx.x;
  if (threadIdx.x != 0) return;
  int k = ws->rem[b];
  long long cum = 0;
  float sumAb = ws->sum_above[b];
  unsigned bsel = 0; int rem = 0; bool found = false;
  for (int bin = NBIN2 - 1; bin >= 0 && !found; --bin) {
    unsigned c = ws->h2_cnt[b][bin];
    if (cum + (long long)c >= (long long)k) { bsel = (unsigned)bin; rem = (int)(k - cum); found = true; }
    else { cum += c; sumAb += ws->h2_sum[b][bin]; }
  }
  ws->b2[b] = bsel; ws->rem[b] = rem; ws->sum_above[b] = sumAb;
}

// Pass 3: histogram bits[9:0] for elements matching top-22 bits.
__global__ void k_pass3(const float* __restrict__ y, const float* __restrict__ o,
                        const float* __restrict__ w, Ws* ws) {
  __shared__ unsigned s_cnt[NBIN3];
  __shared__ float    s_sum[NBIN3];
  const int b   = blockIdx.x / BLOCKS_PER_SAMPLE;
  const int blk = blockIdx.x % BLOCKS_PER_SAMPLE;
  const size_t base = (size_t)b * N_PER + (size_t)blk * EPB;
  const unsigned target22 = (ws->b1[b] << 11) | ws->b2[b];

  for (int i = threadIdx.x; i < NBIN3; i += 256) { s_cnt[i] = 0u; s_sum[i] = 0.0f; }
  __syncthreads();

  for (int i = threadIdx.x; i < EPB; i += 256) {
    size_t idx = base + i;
    float yy = y[idx], oo = o[idx], ww = w[idx];
    float df = oo - yy;
    float mse = df * df;
    float neg = ((oo > 0.0f) && !(ww > 0.0f)) ? mse : 0.0f;
    unsigned bits = __float_as_uint(neg);
    if ((bits >> 10) == target22) {
      unsigned bin = bits & 0x3FFu;
      atomicAdd(&s_cnt[bin], 1u);
      atomicAdd(&s_sum[bin], neg);
    }
  }
  __syncthreads();
  for (int i = threadIdx.x; i < NBIN3; i += 256) {
    unsigned c = s_cnt[i];
    if (c) {
      atomicAdd(&ws->h3_cnt[b][i], c);
      atomicAdd(&ws->h3_sum[b][i], s_sum[i]);
    }
  }
}

__global__ void k_scan3(const float* __restrict__ total_size, Ws* ws) {
  int b = blockIdx.x;
  if (threadIdx.x != 0) return;
  int k = ws->rem[b];
  long long cum = 0;
  float sumAb = ws->sum_above[b];
  unsigned b3 = 0; int remf = 0; bool found = false;
  for (int bin = NBIN3 - 1; bin >= 0 && !found; --bin) {
    unsigned c = ws->h3_cnt[b][bin];
    if (cum + (long long)c >= (long long)k) { b3 = (unsigned)bin; remf = (int)(k - cum); found = true; }
    else { cum += c; sumAb += ws->h3_sum[b][bin]; }
  }
  unsigned tbits = (ws->b1[b] << 21) | (ws->b2[b] << 10) | b3;
  float T = __uint_as_float(tbits);
  float neg_loss = sumAb + ((remf > 0) ? (float)remf * T : 0.0f);
  float ts = total_size[b];
  float safe = (ts > 0.0f) ? ts : 1.0f;
  float per = (ALPHA * ws->pos_loss[b] + neg_loss) / safe;
  ws->per_sample[b] = (ts > 0.0f) ? per : 0.0f;
}

__global__ void k_final(Ws* ws, float* out) {
  // single wave of 32 lanes, one per sample; WMMA all-reduce broadcast
  float v = ws->per_sample[threadIdx.x];
  float total = wave32_allreduce_wmma(v);
  if (threadIdx.x == 0) {
    float train = total / (float)B_SAMPLES;
    float mse_mean = ws->mse_sum / (float)((long long)B_SAMPLES * N_PER);
    out[0] = (train + mse_mean) * 10.0f;
  }
}

// ---------------- launch ----------------
extern "C" void kernel_launch(void* const* d_in, const int* in_sizes, int n_in,
                              void* d_out, int out_size, void* d_ws, size_t ws_size,
                              hipStream_t stream) {
  const float* y  = (const float*)d_in[0];
  const float* o  = (const float*)d_in[1];
  const float* w  = (const float*)d_in[2];
  const float* ts = (const float*)d_in[3];
  Ws* ws = (Ws*)d_ws;
  float* outp = (float*)d_out;

  int nwords = (int)(sizeof(Ws) / sizeof(unsigned));
  k_init<<<256, 256, 0, stream>>>((unsigned*)d_ws, nwords);

  dim3 grid(B_SAMPLES * BLOCKS_PER_SAMPLE);
  k_pass1<<<grid, 256, 0, stream>>>(y, o, w, ws);
  k_scan1<<<B_SAMPLES, 32, 0, stream>>>(ts, ws);
  k_pass2<<<grid, 256, 0, stream>>>(y, o, w, ws);
  k_scan2<<<B_SAMPLES, 32, 0, stream>>>(ws);
  k_pass3<<<grid, 256, 0, stream>>>(y, o, w, ws);
  k_scan3<<<B_SAMPLES, 32, 0, stream>>>(ts, ws);
  k_final<<<1, 32, 0, stream>>>(ws, outp);
}